// GCN_58188216926714
// MI455X (gfx1250) — compile-verified
//
#include <hip/hip_runtime.h>

// GCN 2-layer forward for MI455X (gfx1250, wave32).
// Pipeline: deg -> dinv -> h1=x@W1 (WMMA f32 16x16x4) -> edge scatter (f32 atomics)
//           -> epilogue (self-loop + bias + relu) -> h2=x1@W2 (WMMA) -> scatter -> epilogue.

typedef float v2f __attribute__((ext_vector_type(2)));
typedef float v8f __attribute__((ext_vector_type(8)));

__global__ void zero_f32(float* __restrict__ p, long n) {
  long i = (long)blockIdx.x * blockDim.x + threadIdx.x;
  if (i < n) p[i] = 0.0f;
}

__global__ void degree_kernel(const int* __restrict__ dst, const float* __restrict__ w,
                              float* __restrict__ deg, int E) {
  int e = blockIdx.x * blockDim.x + threadIdx.x;
  if (e < E) atomicAdd(&deg[dst[e]], w[e]);
}

__global__ void dinv_kernel(const float* __restrict__ deg, float* __restrict__ dinv, int n) {
  int i = blockIdx.x * blockDim.x + threadIdx.x;
  if (i < n) dinv[i] = rsqrtf(deg[i] + 1.0f);  // +1 = self-loop weight; always > 0
}

// h[N x 32] = x[N x 64] @ W[64 x 32], one wave per 16-row tile, two 16x16 accumulators.
// A frag (16x4 f32): lane<16 holds row l16, K = k..k+1; lane>=16 holds K = k+2..k+3.
// B frag (4x16 f32): mirrored layout, N = l16 per half, K split by wave half.
__global__ void gemm1_wmma(const float* __restrict__ x, const float* __restrict__ W,
                           float* __restrict__ h, int ntiles) {
  int tile = blockIdx.x * (blockDim.x >> 5) + (threadIdx.x >> 5);
  if (tile >= ntiles) return;                 // wave-uniform: EXEC stays all-1 for WMMA
  const int lane = threadIdx.x & 31;
  const int half = lane >> 4;
  const int l16  = lane & 15;
  const float* xr = x + (long)(tile * 16 + l16) * 64;

  v8f acc0 = {0.f,0.f,0.f,0.f,0.f,0.f,0.f,0.f};
  v8f acc1 = {0.f,0.f,0.f,0.f,0.f,0.f,0.f,0.f};
#pragma unroll
  for (int k = 0; k < 64; k += 4) {
    const int ka = k + half * 2;
    v2f a;  a.x  = xr[ka];                 a.y  = xr[ka + 1];
    v2f b0; b0.x = W[ka * 32 + l16];       b0.y = W[(ka + 1) * 32 + l16];
    v2f b1; b1.x = W[ka * 32 + 16 + l16];  b1.y = W[(ka + 1) * 32 + 16 + l16];
    acc0 = __builtin_amdgcn_wmma_f32_16x16x4_f32(false, a, false, b0, (short)0, acc0, false, false);
    acc1 = __builtin_amdgcn_wmma_f32_16x16x4_f32(false, a, false, b1, (short)0, acc1, false, false);
  }
#pragma unroll
  for (int i = 0; i < 8; ++i) {            // D: VGPR i -> row (i + half*8), col l16
    long m = (long)tile * 16 + i + half * 8;
    h[m * 32 + l16]      = acc0[i];
    h[m * 32 + 16 + l16] = acc1[i];
  }
}

// h2[N x 5] = x1[N x 32] @ W2[32 x 5]; single accumulator, cols >= 5 masked.
__global__ void gemm2_wmma(const float* __restrict__ x1, const float* __restrict__ W,
                           float* __restrict__ h2, int ntiles) {
  int tile = blockIdx.x * (blockDim.x >> 5) + (threadIdx.x >> 5);
  if (tile >= ntiles) return;
  const int lane = threadIdx.x & 31;
  const int half = lane >> 4;
  const int l16  = lane & 15;
  const bool colok = (l16 < 5);
  const float* xr = x1 + (long)(tile * 16 + l16) * 32;

  v8f acc = {0.f,0.f,0.f,0.f,0.f,0.f,0.f,0.f};
#pragma unroll
  for (int k = 0; k < 32; k += 4) {
    const int ka = k + half * 2;
    v2f a; a.x = xr[ka];                          a.y = xr[ka + 1];
    v2f b; b.x = colok ? W[ka * 5 + l16] : 0.0f;  b.y = colok ? W[(ka + 1) * 5 + l16] : 0.0f;
    acc = __builtin_amdgcn_wmma_f32_16x16x4_f32(false, a, false, b, (short)0, acc, false, false);
  }
#pragma unroll
  for (int i = 0; i < 8; ++i) {
    long m = (long)tile * 16 + i + half * 8;
    if (colok) h2[m * 5 + l16] = acc[i];
  }
}

// One thread per (edge, feature): coalesced gather of h[src*32..] and coalesced atomics.
__global__ void scatter32(const int* __restrict__ src, const int* __restrict__ dst,
                          const float* __restrict__ w, const float* __restrict__ dinv,
                          const float* __restrict__ h, float* __restrict__ out, int E) {
  long g = (long)blockIdx.x * blockDim.x + threadIdx.x;
  long e = g >> 5;
  if (e >= E) return;
  int j = (int)(g & 31);
  int s = src[e], d = dst[e];
  float nrm = dinv[s] * w[e] * dinv[d];
  atomicAdd(&out[(long)d * 32 + j], h[(long)s * 32 + j] * nrm);
}

// 8 lanes per edge, 5 active (keeps addressing power-of-two aligned).
__global__ void scatter5(const int* __restrict__ src, const int* __restrict__ dst,
                         const float* __restrict__ w, const float* __restrict__ dinv,
                         const float* __restrict__ h, float* __restrict__ out, int E) {
  long g = (long)blockIdx.x * blockDim.x + threadIdx.x;
  long e = g >> 3;
  if (e >= E) return;
  int j = (int)(g & 7);
  if (j >= 5) return;
  int s = src[e], d = dst[e];
  float nrm = dinv[s] * w[e] * dinv[d];
  atomicAdd(&out[(long)d * 5 + j], h[(long)s * 5 + j] * nrm);
}

// x1 = relu(agg + dinv^2 * h1 + b1)  (self-loop term fused, no atomics needed)
__global__ void epilogue1(float* __restrict__ x1, const float* __restrict__ h1,
                          const float* __restrict__ dinv, const float* __restrict__ b, int N) {
  long g = (long)blockIdx.x * blockDim.x + threadIdx.x;
  if (g >= (long)N * 32) return;
  int i = (int)(g >> 5);
  int j = (int)(g & 31);
  float di = dinv[i];
  float v = x1[g] + di * di * h1[g] + b[j];
  x1[g] = fmaxf(v, 0.0f);
}

// x2 = agg + dinv^2 * h2 + b2  (no relu on output layer)
__global__ void epilogue2(float* __restrict__ x2, const float* __restrict__ h2,
                          const float* __restrict__ dinv, const float* __restrict__ b, int N) {
  long g = (long)blockIdx.x * blockDim.x + threadIdx.x;
  if (g >= (long)N * 5) return;
  int i = (int)(g / 5);
  int j = (int)(g % 5);
  float di = dinv[i];
  x2[g] = x2[g] + di * di * h2[g] + b[j];
}

static inline int cdiv(long a, long b) { return (int)((a + b - 1) / b); }

extern "C" void kernel_launch(void* const* d_in, const int* in_sizes, int n_in,
                              void* d_out, int out_size, void* d_ws, size_t ws_size,
                              hipStream_t stream) {
  const float* x  = (const float*)d_in[0];   // [N, 64]
  const int*   ei = (const int*)d_in[1];     // [2, E]
  const float* ew = (const float*)d_in[2];   // [E]
  const float* W1 = (const float*)d_in[3];   // [64, 32]
  const float* b1 = (const float*)d_in[4];   // [32]
  const float* W2 = (const float*)d_in[5];   // [32, 5]
  const float* b2 = (const float*)d_in[6];   // [5]

  const int N = in_sizes[0] / 64;            // 100000
  const int E = in_sizes[2];                 // 3200000
  const int* src = ei;
  const int* dst = ei + E;

  float* out = (float*)d_out;
  float* x2  = out;                          // [N, 5]  (tuple element 0)
  float* x1  = out + (long)N * 5;            // [N, 32] (tuple element 1)

  float* ws   = (float*)d_ws;                // 39*N floats ~ 15.6 MB
  float* deg  = ws;                          // [N]
  float* dinv = ws + N;                      // [N]
  float* h1   = ws + 2L * N;                 // [N, 32]
  float* h2   = ws + 34L * N;                // [N, 5]

  const int ntiles = N / 16;                 // 6250, exact

  zero_f32<<<cdiv(N, 256), 256, 0, stream>>>(deg, N);
  zero_f32<<<cdiv((long)N * 37, 256), 256, 0, stream>>>(out, (long)N * 37);

  degree_kernel<<<cdiv(E, 256), 256, 0, stream>>>(dst, ew, deg, E);
  dinv_kernel<<<cdiv(N, 256), 256, 0, stream>>>(deg, dinv, N);

  gemm1_wmma<<<cdiv(ntiles, 4), 128, 0, stream>>>(x, W1, h1, ntiles);
  scatter32<<<cdiv((long)E * 32, 256), 256, 0, stream>>>(src, dst, ew, dinv, h1, x1, E);
  epilogue1<<<cdiv((long)N * 32, 256), 256, 0, stream>>>(x1, h1, dinv, b1, N);

  gemm2_wmma<<<cdiv(ntiles, 4), 128, 0, stream>>>(x1, W2, h2, ntiles);
  scatter5<<<cdiv((long)E * 8, 256), 256, 0, stream>>>(src, dst, ew, dinv, h2, x2, E);
  epilogue2<<<cdiv((long)N * 5, 256), 256, 0, stream>>>(x2, h2, dinv, b2, N);
}